// ConRelEncoder_13237089206715
// MI455X (gfx1250) — compile-verified
//
#include <hip/hip_runtime.h>
#include <hip/hip_bf16.h>

#define HD 128
#define TOPK 5
#define NRES_MAX 2048

typedef __attribute__((ext_vector_type(2))) float v2f;
typedef __attribute__((ext_vector_type(8))) float v8f;

// ---------------- zero-init workspace ----------------
__global__ void zero_kernel(float* feat_sum, int* in_deg, int* out_deg, int nNodes) {
    int i = blockIdx.x * blockDim.x + threadIdx.x;
    long long total = (long long)nNodes * HD;
    if ((long long)i < total) feat_sum[i] = 0.0f;
    if (i < nNodes) { in_deg[i] = 0; out_deg[i] = 0; }
}

// ---------------- l2-normalize res_ent_emb -> z2 ----------------
__global__ void norm_res_kernel(const float* __restrict__ res, float* __restrict__ z2) {
    int j = blockIdx.x;
    int t = threadIdx.x;            // 128 threads
    __shared__ float s[HD];
    float x = res[(long long)j * HD + t];
    s[t] = x * x;
    __syncthreads();
    for (int st = 64; st > 0; st >>= 1) {
        if (t < st) s[t] += s[t + st];
        __syncthreads();
    }
    float nrm = fmaxf(sqrtf(s[0]), 1e-12f);
    z2[(long long)j * HD + t] = x / nrm;
}

// ---------------- degree accumulation ----------------
__global__ void degree_kernel(const int* __restrict__ src, const int* __restrict__ dst,
                              int* in_deg, int* out_deg, int nE) {
    int e = blockIdx.x * blockDim.x + threadIdx.x;
    if (e < nE) {
        atomicAdd(&in_deg[dst[e]], 1);
        atomicAdd(&out_deg[src[e]], 1);
    }
}

// ---------------- edge feature scatter-add ----------------
// one edge -> 32 threads, each handles 4 contiguous floats (float4 load, 4 f32 atomics)
__global__ void scatter_kernel(const int* __restrict__ dst, const int* __restrict__ rel,
                               const int* __restrict__ inv,
                               const float* __restrict__ rel_head,
                               const float* __restrict__ rel_tail,
                               float* feat_sum, long long nWork) {
    long long tid = (long long)blockIdx.x * blockDim.x + threadIdx.x;
    if (tid >= nWork) return;
    int e    = (int)(tid >> 5);
    int lane = (int)(tid & 31);
    int d  = dst[e];
    int r  = rel[e];
    int iv = inv[e];
    const float* srow = (iv == 1 ? rel_head : rel_tail) + (long long)r * HD + lane * 4;
    float* drow = feat_sum + (long long)d * HD + lane * 4;
    float4 v = *(const float4*)srow;
    atomicAdd(drow + 0, v.x);
    atomicAdd(drow + 1, v.y);
    atomicAdd(drow + 2, v.z);
    atomicAdd(drow + 3, v.w);
}

// ---------------- finalize: mean-divide or zero-degree top-k fallback ----------------
__global__ void finalize_kernel(const float* __restrict__ feat_sum,
                                const int* __restrict__ in_deg, const int* __restrict__ out_deg,
                                const float* __restrict__ ent_type,
                                const float* __restrict__ z2,
                                const float* __restrict__ res_ent,
                                float* __restrict__ feat, int nres) {
    int n = blockIdx.x;
    int t = threadIdx.x;            // 128 threads
    int id = in_deg[n], od = out_deg[n];
    if (id + od != 0) {             // block-uniform branch
        float d = (float)(id > 1 ? id : 1);
        feat[(long long)n * HD + t] = feat_sum[(long long)n * HD + t] / d;
        return;
    }
    // zero-degree fallback: sim row vs all res entities, top-5, softmax(T=0.2)
    __shared__ float s_x[HD];
    __shared__ float s_sim[NRES_MAX];
    __shared__ float s_rv[HD];
    __shared__ int   s_ri[HD];
    __shared__ float s_topv[TOPK];
    __shared__ int   s_topi[TOPK];

    float x = ent_type[(long long)n * HD + t];
    s_x[t] = x * x;
    __syncthreads();
    for (int st = 64; st > 0; st >>= 1) {
        if (t < st) s_x[t] += s_x[t + st];
        __syncthreads();
    }
    float nrm = fmaxf(sqrtf(s_x[0]), 1e-12f);
    __syncthreads();
    s_x[t] = x / nrm;               // normalized query row in LDS
    __syncthreads();

    for (int j = t; j < nres; j += HD) {
        const float* zr = z2 + (long long)j * HD;
        float acc = 0.0f;
        #pragma unroll 8
        for (int k = 0; k < HD; ++k) acc += s_x[k] * zr[k];
        s_sim[j] = acc;
    }
    __syncthreads();

    for (int kk = 0; kk < TOPK; ++kk) {
        float best = -3.4e38f; int bi = 0x7fffffff;
        for (int j = t; j < nres; j += HD) {
            float v = s_sim[j];
            if (v > best || (v == best && j < bi)) { best = v; bi = j; }
        }
        s_rv[t] = best; s_ri[t] = bi;
        __syncthreads();
        for (int st = 64; st > 0; st >>= 1) {
            if (t < st) {
                if (s_rv[t + st] > s_rv[t] ||
                    (s_rv[t + st] == s_rv[t] && s_ri[t + st] < s_ri[t])) {
                    s_rv[t] = s_rv[t + st]; s_ri[t] = s_ri[t + st];
                }
            }
            __syncthreads();
        }
        if (t == 0) {
            s_topv[kk] = s_rv[0];
            s_topi[kk] = s_ri[0];
            s_sim[s_ri[0]] = -3.4e38f;   // knock out the winner
        }
        __syncthreads();
    }

    float mx = s_topv[0];            // selected in descending order
    float w[TOPK]; float wsum = 0.0f;
    #pragma unroll
    for (int kk = 0; kk < TOPK; ++kk) { w[kk] = expf((s_topv[kk] - mx) * 5.0f); wsum += w[kk]; }
    float acc = 0.0f;
    #pragma unroll
    for (int kk = 0; kk < TOPK; ++kk)
        acc += (w[kk] / wsum) * res_ent[(long long)s_topi[kk] * HD + t];
    feat[(long long)n * HD + t] = acc;
}

// ---------------- WMMA f32 MLP: out = relu(feat@W1^T+b1)@W2^T+b2 ----------------
// 4 waves/block, one 16-row tile per wave. Per layer: 8 col-tiles x 32 K-blocks
// of V_WMMA_F32_16X16X4_F32. Layer-1 result staged via LDS (C/D -> A layout).
__global__ __launch_bounds__(128) void mlp_wmma_kernel(
        const float* __restrict__ feat,
        const float* __restrict__ W1, const float* __restrict__ b1,
        const float* __restrict__ W2, const float* __restrict__ b2,
        float* __restrict__ out, int nTiles, int nNodes) {
    __shared__ float lds_h[4][16 * HD];       // 8KB per wave
    const int wave = threadIdx.x >> 5;
    const int lane = threadIdx.x & 31;
    const int tileRaw = blockIdx.x * 4 + wave;
    const bool active = (tileRaw < nTiles);   // wave-uniform
    const int tile = active ? tileRaw : (nTiles - 1);
    const int rowBase = tile * 16;

    const int mr    = lane & 15;              // A: row-in-tile; B/C/D: col-in-tile
    const int khalf = (lane >> 4) * 2;        // 0 or 2 (K sub-offset)
    const int mOff  = (lane >> 4) * 8;        // C/D: M offset for lanes 16..31

    // ---- load A fragments (feat tile) ----
    v2f a[32];
    {
        int r = rowBase + mr; if (r >= nNodes) r = nNodes - 1;
        const float* arow = feat + (long long)r * HD + khalf;
        #pragma unroll
        for (int kb = 0; kb < 32; ++kb) {
            a[kb].x = arow[kb * 4 + 0];
            a[kb].y = arow[kb * 4 + 1];
        }
    }

    // ---- layer 1 ----
    float* hl = &lds_h[wave][0];
    #pragma unroll 2
    for (int nt = 0; nt < 8; ++nt) {
        v8f acc = {};
        const float* wrow = W1 + (long long)(nt * 16 + mr) * HD + khalf;
        #pragma unroll
        for (int kb = 0; kb < 32; ++kb) {
            v2f b; b.x = wrow[kb * 4 + 0]; b.y = wrow[kb * 4 + 1];
            acc = __builtin_amdgcn_wmma_f32_16x16x4_f32(
                      false, a[kb], false, b, (short)0, acc, false, false);
        }
        float bias = b1[nt * 16 + mr];
        #pragma unroll
        for (int v = 0; v < 8; ++v) {
            float val = acc[v] + bias;
            val = val > 0.0f ? val : 0.0f;                // ReLU
            hl[(mOff + v) * HD + nt * 16 + mr] = val;     // C/D layout -> row-major LDS
        }
    }
    __syncthreads();

    // ---- reload hidden as A fragments ----
    {
        const float* hrow = hl + mr * HD + khalf;
        #pragma unroll
        for (int kb = 0; kb < 32; ++kb) {
            a[kb].x = hrow[kb * 4 + 0];
            a[kb].y = hrow[kb * 4 + 1];
        }
    }

    // ---- layer 2 ----
    #pragma unroll 2
    for (int nt = 0; nt < 8; ++nt) {
        v8f acc = {};
        const float* wrow = W2 + (long long)(nt * 16 + mr) * HD + khalf;
        #pragma unroll
        for (int kb = 0; kb < 32; ++kb) {
            v2f b; b.x = wrow[kb * 4 + 0]; b.y = wrow[kb * 4 + 1];
            acc = __builtin_amdgcn_wmma_f32_16x16x4_f32(
                      false, a[kb], false, b, (short)0, acc, false, false);
        }
        if (active) {
            float bias = b2[nt * 16 + mr];
            #pragma unroll
            for (int v = 0; v < 8; ++v) {
                int row = rowBase + mOff + v;
                if (row < nNodes)
                    out[(long long)row * HD + nt * 16 + mr] = acc[v] + bias;
            }
        }
    }
}

extern "C" void kernel_launch(void* const* d_in, const int* in_sizes, int n_in,
                              void* d_out, int out_size, void* d_ws, size_t ws_size,
                              hipStream_t stream) {
    const int*   src      = (const int*)  d_in[0];
    const int*   dst      = (const int*)  d_in[1];
    const int*   rel      = (const int*)  d_in[2];
    const int*   inv      = (const int*)  d_in[3];
    const float* ent_type = (const float*)d_in[4];
    const float* rel_head = (const float*)d_in[5];
    const float* rel_tail = (const float*)d_in[6];
    const float* res_ent  = (const float*)d_in[7];
    const float* W1       = (const float*)d_in[8];
    const float* b1       = (const float*)d_in[9];
    const float* W2       = (const float*)d_in[10];
    const float* b2       = (const float*)d_in[11];
    float* out = (float*)d_out;

    const int E    = in_sizes[0];
    const int N    = in_sizes[4] / HD;
    const int nres = in_sizes[7] / HD;

    // workspace carve-up
    char* ws = (char*)d_ws;
    size_t off = 0;
    float* feat_sum = (float*)(ws + off); off += (size_t)N * HD * sizeof(float);
    float* feat     = (float*)(ws + off); off += (size_t)N * HD * sizeof(float);
    float* z2       = (float*)(ws + off); off += (size_t)nres * HD * sizeof(float);
    int*   in_deg   = (int*)  (ws + off); off += (size_t)N * sizeof(int);
    int*   out_deg  = (int*)  (ws + off); off += (size_t)N * sizeof(int);
    (void)ws_size; (void)n_in; (void)out_size;

    // 1. zero feat_sum + degrees
    {
        long long total = (long long)N * HD;
        int blocks = (int)((total + 255) / 256);
        zero_kernel<<<blocks, 256, 0, stream>>>(feat_sum, in_deg, out_deg, N);
    }
    // 2. normalized res embeddings
    norm_res_kernel<<<nres, HD, 0, stream>>>(res_ent, z2);
    // 3. degrees
    degree_kernel<<<(E + 255) / 256, 256, 0, stream>>>(src, dst, in_deg, out_deg, E);
    // 4. edge scatter-add
    {
        long long nWork = (long long)E * 32;
        int blocks = (int)((nWork + 255) / 256);
        scatter_kernel<<<blocks, 256, 0, stream>>>(dst, rel, inv, rel_head, rel_tail,
                                                   feat_sum, nWork);
    }
    // 5. finalize (mean or zero-degree fallback)
    finalize_kernel<<<N, HD, 0, stream>>>(feat_sum, in_deg, out_deg, ent_type,
                                          z2, res_ent, feat, nres);
    // 6. WMMA MLP
    {
        int nTiles = (N + 15) / 16;
        int blocks = (nTiles + 3) / 4;
        mlp_wmma_kernel<<<blocks, 128, 0, stream>>>(feat, W1, b1, W2, b2, out, nTiles, N);
    }
}